// PositionEmbeddingSine_35390530519696
// MI455X (gfx1250) — compile-verified
//
#include <hip/hip_runtime.h>
#include <hip/hip_bf16.h>
#include <cstdint>
#include <cstddef>

// PositionEmbeddingSine for MI455X (gfx1250, wave32).
//
// Layout per row (192 floats):
//   for c in {x,y,z}: for k in 0..31: [ sin(v_c * 10000^(-k/32)),
//                                       cos(v_c * 10000^(-k/32)) ]
// where v_c = coord_c / (S_c - 1 + 1e-6). The reference's 2*pi SCALE cancels
// against v_sin_f32/v_cos_f32 taking their argument in revolutions.
//
// One wave per row, lane == pair index k. Each lane: 1x v_exp_f32,
// 3x (v_sin_f32 + v_cos_f32), 3x coalesced non-temporal 8-byte stores
// (global_store_b64 th:TH_STORE_NT — output is write-once streaming, keep it
// out of the 192MB L2). Coords for the block's 8 rows are staged into LDS
// with the CDNA5 async global->LDS path (ASYNCcnt) and broadcast to all
// 8 waves via a single ds_load_b128 per wave.

typedef __attribute__((ext_vector_type(2))) float v2f;   // native clang vector

#define PAIRS            32            // feature pairs per coordinate
#define ROWS_PER_BLOCK   8
#define BLOCK_THREADS    (ROWS_PER_BLOCK * 32)
#define LOG2_10000       13.287712379549449f

// ---------------------------------------------------------------------------
// Main embedding kernel (first in file so the disasm snippet shows it).
// ---------------------------------------------------------------------------
__global__ __launch_bounds__(BLOCK_THREADS)
void pes_main(const int* __restrict__ coords,
              const int* __restrict__ starts,
              const int* __restrict__ sx_p, const int* __restrict__ sy_p,
              const int* __restrict__ sz_p, const int* __restrict__ nb_p,
              const int* __restrict__ ml_p,
              float* __restrict__ out, int M)
{
    __shared__ int smem[ROWS_PER_BLOCK * 4];

    const int       tid  = (int)threadIdx.x;
    const int       lane = tid & 31;
    const int       wid  = tid >> 5;
    const long long row0 = (long long)blockIdx.x * ROWS_PER_BLOCK;

    // Stage this block's coords (8 rows x 4 ints = 128 B) into LDS via the
    // CDNA5 async global->LDS copy (tracked with ASYNCcnt). Wave 0 issues
    // 32 b32 async loads, drains its ASYNCcnt, then the workgroup barrier
    // makes the LDS data visible to all 8 waves.
    if (tid < ROWS_PER_BLOCK * 4) {                 // exactly wave 0 (wave32)
        long long g = row0 * 4 + tid;
        if (g < (long long)M * 4) {
            unsigned   lds_addr = (unsigned)(uintptr_t)(&smem[tid]);
            const int* gptr     = coords + g;
            asm volatile("global_load_async_to_lds_b32 %0, %1, off"
                         :: "v"(lds_addr), "v"(gptr) : "memory");
        }
        asm volatile("s_wait_asynccnt 0" ::: "memory");
    }
    __syncthreads();

    const long long r = row0 + wid;
    if (r >= M) return;

    // Per-wave uniform row data from LDS (single ds_load_b128 broadcast).
    const int   b  = smem[wid * 4 + 0];
    const float xv = (float)smem[wid * 4 + 1];
    const float yv = (float)smem[wid * 4 + 2];
    const float zv = (float)smem[wid * 4 + 3];

    const int nb = *nb_p;
    const int ml = *ml_p;
    if (b < 0 || b >= nb) return;
    const int slot = (int)r - starts[b];
    if (slot < 0 || slot >= ml) return;

    // Normalization (SCALE = 2*pi folded into the hw sin/cos revolutions).
    const float inv_dx = 1.0f / ((float)(*sx_p - 1) + 1e-6f);
    const float inv_dy = 1.0f / ((float)(*sy_p - 1) + 1e-6f);
    const float inv_dz = 1.0f / ((float)(*sz_p - 1) + 1e-6f);
    const float n0 = xv * inv_dx;
    const float n1 = yv * inv_dy;
    const float n2 = zv * inv_dz;

    // inv_t = 10000^(-lane/32) = exp2(-lane * log2(10000)/32)  (v_exp_f32)
    const float inv_t =
        __builtin_amdgcn_exp2f((float)lane * (-LOG2_10000 / (float)PAIRS));

    const float a0 = n0 * inv_t;   // arguments in revolutions, in [0, 1]
    const float a1 = n1 * inv_t;
    const float a2 = n2 * inv_t;

    v2f* __restrict__ orow =
        (v2f*)(out + ((size_t)b * (size_t)ml + (size_t)slot) * (size_t)(6 * PAIRS));
    v2f* __restrict__ o = orow + lane;

    // Streaming output: non-temporal b64 stores, fully coalesced per wave.
    v2f p0 = { __builtin_amdgcn_sinf(a0), __builtin_amdgcn_cosf(a0) };
    v2f p1 = { __builtin_amdgcn_sinf(a1), __builtin_amdgcn_cosf(a1) };
    v2f p2 = { __builtin_amdgcn_sinf(a2), __builtin_amdgcn_cosf(a2) };
    __builtin_nontemporal_store(p0, o + 0 * PAIRS);
    __builtin_nontemporal_store(p1, o + 1 * PAIRS);
    __builtin_nontemporal_store(p2, o + 2 * PAIRS);
}

// ---------------------------------------------------------------------------
// Batch start offsets (lower_bound of each batch id in sorted b).
// slot(i) = i - starts[b[i]] reproduces  i - searchsorted(b, b, 'left').
// ---------------------------------------------------------------------------
__global__ void pes_batch_starts(const int* __restrict__ coords, int M,
                                 const int* __restrict__ nb_p,
                                 int* __restrict__ starts)
{
    int v  = (int)threadIdx.x;
    int nb = *nb_p;
    if (v >= nb) return;
    int lo = 0, hi = M;
    while (lo < hi) {
        int mid = (lo + hi) >> 1;
        if (coords[(size_t)mid * 4] < v) lo = mid + 1;
        else                             hi = mid;
    }
    starts[v] = lo;
}

// ---------------------------------------------------------------------------
// Launcher. Inputs (setup_inputs order):
//   d_in[0] coords (M*4 int32), d_in[1] sx, d_in[2] sy, d_in[3] sz,
//   d_in[4] n_batches, d_in[5] max_len   (scalars as 1-elem int arrays)
// d_out: n_batches * max_len * 192 float32.
// d_ws:  batch-start offsets (int per batch).
// ---------------------------------------------------------------------------
extern "C" void kernel_launch(void* const* d_in, const int* in_sizes, int n_in,
                              void* d_out, int out_size, void* d_ws, size_t ws_size,
                              hipStream_t stream)
{
    const int* coords = (const int*)d_in[0];
    const int* sx_p   = (const int*)d_in[1];
    const int* sy_p   = (const int*)d_in[2];
    const int* sz_p   = (const int*)d_in[3];
    const int* nb_p   = (const int*)d_in[4];
    const int* ml_p   = (const int*)d_in[5];
    float*     out    = (float*)d_out;
    int*       starts = (int*)d_ws;

    const int M = in_sizes[0] / 4;

    // Reference starts from jnp.zeros; rows not covered by the scatter must
    // be zero. ~252 MB @ 23.3 TB/s ~= 11 us.
    (void)hipMemsetAsync(d_out, 0, (size_t)out_size * sizeof(float), stream);

    pes_batch_starts<<<1, 256, 0, stream>>>(coords, M, nb_p, starts);

    const int nblocks = (M + ROWS_PER_BLOCK - 1) / ROWS_PER_BLOCK;
    pes_main<<<nblocks, BLOCK_THREADS, 0, stream>>>(
        coords, starts, sx_p, sy_p, sz_p, nb_p, ml_p, out, M);
}